// SingleGnnLayerwithVirtualNode_17669495456466
// MI455X (gfx1250) — compile-verified
//
#include <hip/hip_runtime.h>

#define IN_DIM 128
#define EMB    128
#define HID    256
#define BN_EPS 1e-5f

typedef __attribute__((ext_vector_type(2))) float v2f;
typedef __attribute__((ext_vector_type(4))) float v4f;
typedef __attribute__((ext_vector_type(8))) float v8f;

__device__ __forceinline__ v8f wmma4(v2f a, v2f b, v8f c) {
  // D = A(16x4 f32) * B(4x16 f32) + C(16x16 f32), exact fp32 on the WMMA pipe
  return __builtin_amdgcn_wmma_f32_16x16x4_f32(
      /*neg_a=*/false, a, /*neg_b=*/false, b,
      /*c_mod=*/(short)0, c, /*reuse_a=*/false, /*reuse_b=*/false);
}

// A fragment: rows r0..r0+15, K-slice [k0,k0+4). lane<16 -> K={k0,k0+1}, lane>=16 -> K={k0+2,k0+3}
__device__ __forceinline__ v2f load_a(const float* __restrict__ A, int lda,
                                      int r0, int k0, int lsub, int half) {
  const float* p = A + (size_t)(r0 + lsub) * lda + (k0 + 2 * half);
  return *(const v2f*)p;  // contiguous pair -> global_load_b64
}

// Packed-B fragment from global: one coalesced global_load_b64
__device__ __forceinline__ v2f load_bp(const float* __restrict__ P, int nT,
                                       int ks, int t, int lane) {
  return *(const v2f*)(P + ((size_t)(ks * nT + t) * 32 + lane) * 2);
}

// Packed-B fragment from LDS: lane-contiguous pair -> conflict-free ds_load_b64
__device__ __forceinline__ v2f load_bl(const float* lw, int nT,
                                       int ks, int t, int lane) {
  return *(const v2f*)(lw + ((size_t)(ks * nT + t) * 32 + lane) * 2);
}

// Cooperative global->LDS stage, 16B vectors
__device__ __forceinline__ void stage_lds(const float* __restrict__ src,
                                          float* dst, int nfloats,
                                          int tid, int nthreads) {
  for (int i = tid * 4; i < nfloats; i += nthreads * 4)
    *(v4f*)(dst + i) = *(const v4f*)(src + i);
}

// Repack row-major W[K,N] into fragment-major layout:
// P[ks][t][lane][j] = W[(4*ks + 2*(lane>=16) + j) * N + 16*t + (lane&15)]
__global__ void pack_w_kernel(const float* __restrict__ W, float* __restrict__ P,
                              int K, int Ncols) {
  const int nT = Ncols >> 4;
  const int total = (K >> 2) * nT * 32;
  const int idx = blockIdx.x * blockDim.x + threadIdx.x;
  if (idx >= total) return;
  const int lane = idx & 31;
  const int t = (idx >> 5) % nT;
  const int ks = idx / (32 * nT);
  const int half = lane >> 4, lsub = lane & 15;
  const int col = t * 16 + lsub;
  const int krow = ks * 4 + 2 * half;
  v2f v;
  v.x = W[(size_t)krow * Ncols + col];
  v.y = W[(size_t)(krow + 1) * Ncols + col];
  *(v2f*)(P + (size_t)idx * 2) = v;
}

__global__ void zero_f32_kernel(float* __restrict__ p, unsigned long long n) {
  unsigned long long i = (unsigned long long)blockIdx.x * blockDim.x + threadIdx.x;
  if (i < n) p[i] = 0.0f;
}

// h0 = x @ feat_W + feat_b + vn_emb[0]   (vn[batch] == vn_emb row 0 for every node)
// 8 waves/block share feat_W in LDS; each wave does 16 rows.
__global__ __launch_bounds__(256) void encode_kernel(
    const float* __restrict__ x, const float* __restrict__ Wp,
    const float* __restrict__ bias, const float* __restrict__ vn,
    float* __restrict__ h0, int N) {
  __shared__ __align__(16) float wl[IN_DIM * EMB];
  const int tid = threadIdx.x;
  stage_lds(Wp, wl, IN_DIM * EMB, tid, 256);
  __syncthreads();

  const int lane = tid & 31;
  const int wave = tid >> 5;
  const int half = lane >> 4, lsub = lane & 15;
  const int r0 = blockIdx.x * (8 * 16) + wave * 16;
  if (r0 >= N) return;  // uniform per wave: EXEC stays all-ones for WMMA

  v8f acc[8] = {};
  for (int ks = 0; ks < IN_DIM / 4; ++ks) {
    v2f a = load_a(x, IN_DIM, r0, ks * 4, lsub, half);
#pragma unroll
    for (int t = 0; t < 8; ++t) {
      v2f b = load_bl(wl, 8, ks, t, lane);
      acc[t] = wmma4(a, b, acc[t]);
    }
  }
#pragma unroll
  for (int t = 0; t < 8; ++t) {
    const int col = t * 16 + lsub;
    const float add = bias[col] + vn[col];
#pragma unroll
    for (int i = 0; i < 8; ++i) {
      const int row = r0 + 8 * half + i;
      h0[(size_t)row * EMB + col] = acc[t][i] + add;
    }
  }
}

// edge_emb = edge_attr @ edge_W + edge_b ; msg = relu(h0[src] + edge_emb) ; aggr[dst] += msg
// 8 waves/block share edge_W in LDS; each wave does 32 edges (2 M-tiles, B reused 2x).
__global__ __launch_bounds__(256) void edge_kernel(
    const float* __restrict__ edge_attr, const float* __restrict__ Wp,
    const float* __restrict__ bias, const int* __restrict__ src,
    const int* __restrict__ dst, const float* __restrict__ h0,
    float* __restrict__ aggr, int E) {
  __shared__ __align__(16) float wl[IN_DIM * EMB];
  const int tid = threadIdx.x;
  stage_lds(Wp, wl, IN_DIM * EMB, tid, 256);
  __syncthreads();

  const int lane = tid & 31;
  const int wave = tid >> 5;
  const int half = lane >> 4, lsub = lane & 15;
  const int e0 = blockIdx.x * (8 * 32) + wave * 32;
  if (e0 >= E) return;  // uniform per wave (E % 32 == 0)

  v8f acc[16] = {};
  for (int ks = 0; ks < IN_DIM / 4; ++ks) {
    v2f a0 = load_a(edge_attr, IN_DIM, e0,      ks * 4, lsub, half);
    v2f a1 = load_a(edge_attr, IN_DIM, e0 + 16, ks * 4, lsub, half);
#pragma unroll
    for (int t = 0; t < 8; ++t) {
      v2f b = load_bl(wl, 8, ks, t, lane);
      acc[t]     = wmma4(a0, b, acc[t]);
      acc[8 + t] = wmma4(a1, b, acc[8 + t]);
    }
  }

  int s_i[16], d_i[16];
#pragma unroll
  for (int i = 0; i < 8; ++i) {
    const int ea = e0 + 8 * half + i;
    s_i[i] = src[ea];
    d_i[i] = dst[ea];
    const int eb2 = ea + 16;
    s_i[8 + i] = src[eb2];
    d_i[8 + i] = dst[eb2];
  }

#pragma unroll
  for (int t = 0; t < 8; ++t) {
    const int col = t * 16 + lsub;
    const float eb = bias[col];
#pragma unroll
    for (int i = 0; i < 8; ++i) {
      float v0 = acc[t][i] + eb + h0[(size_t)s_i[i] * EMB + col];
      unsafeAtomicAdd(&aggr[(size_t)d_i[i] * EMB + col], fmaxf(v0, 0.0f));
      float v1 = acc[8 + t][i] + eb + h0[(size_t)s_i[8 + i] * EMB + col];
      unsafeAtomicAdd(&aggr[(size_t)d_i[8 + i] * EMB + col], fmaxf(v1, 0.0f));
    }
  }
}

// Fused MLP: pre=(1+eps)*h0+aggr ; z=relu(bn1(pre@W1+b1)) -> LDS ;
// h=z@W2+b2 ; out = relu(bn(h)) + h0   (in place over h0 buffer)
__global__ __launch_bounds__(32) void mlp_kernel(
    float* __restrict__ h0_out, const float* __restrict__ aggr,
    const float* __restrict__ eps_p,
    const float* __restrict__ W1p, const float* __restrict__ b1,
    const float* __restrict__ bn1_g, const float* __restrict__ bn1_b,
    const float* __restrict__ bn1_m, const float* __restrict__ bn1_v,
    const float* __restrict__ W2p, const float* __restrict__ b2,
    const float* __restrict__ bn_g, const float* __restrict__ bn_b,
    const float* __restrict__ bn_m, const float* __restrict__ bn_v) {
  __shared__ float zs[16 * 258];  // stride 258: 8B-aligned b64 LDS loads, conflict-free
  const int lane = threadIdx.x & 31;
  const int half = lane >> 4, lsub = lane & 15;
  const int r0 = blockIdx.x * 16;
  const float epl = 1.0f + eps_p[0];

  // ---- GEMM1: pre[16,128] @ W1[128,256], pre computed on the fly ----
  v8f acc1[16] = {};
  for (int ks = 0; ks < EMB / 4; ++ks) {
    const int row = r0 + lsub;
    const size_t base = (size_t)row * EMB + (ks * 4 + 2 * half);
    v2f a;
    a.x = epl * h0_out[base]     + aggr[base];
    a.y = epl * h0_out[base + 1] + aggr[base + 1];
#pragma unroll
    for (int t = 0; t < 16; ++t) {
      v2f b = load_bp(W1p, 16, ks, t, lane);
      acc1[t] = wmma4(a, b, acc1[t]);
    }
  }

  // ---- BN1 + ReLU, stash z into LDS ----
#pragma unroll
  for (int t = 0; t < 16; ++t) {
    const int col = t * 16 + lsub;
    const float s = bn1_g[col] * rsqrtf(bn1_v[col] + BN_EPS);
    const float o = (b1[col] - bn1_m[col]) * s + bn1_b[col];
#pragma unroll
    for (int i = 0; i < 8; ++i) {
      zs[(8 * half + i) * 258 + col] = fmaxf(acc1[t][i] * s + o, 0.0f);
    }
  }
  __syncthreads();

  // ---- GEMM2: z[16,256] (LDS) @ W2[256,128] ----
  v8f acc2[8] = {};
  for (int ks = 0; ks < HID / 4; ++ks) {
    const v2f a = *(const v2f*)&zs[lsub * 258 + (ks * 4 + 2 * half)];  // ds_load_b64
#pragma unroll
    for (int t = 0; t < 8; ++t) {
      v2f b = load_bp(W2p, 8, ks, t, lane);
      acc2[t] = wmma4(a, b, acc2[t]);
    }
  }

  // ---- BN + ReLU + residual, in-place ----
#pragma unroll
  for (int t = 0; t < 8; ++t) {
    const int col = t * 16 + lsub;
    const float s = bn_g[col] * rsqrtf(bn_v[col] + BN_EPS);
    const float o = (b2[col] - bn_m[col]) * s + bn_b[col];
#pragma unroll
    for (int i = 0; i < 8; ++i) {
      const int row = r0 + 8 * half + i;
      const size_t idx = (size_t)row * EMB + col;
      const float h0v = h0_out[idx];               // read residual before overwrite
      h0_out[idx] = fmaxf(acc2[t][i] * s + o, 0.0f) + h0v;
    }
  }
}

extern "C" void kernel_launch(void* const* d_in, const int* in_sizes, int n_in,
                              void* d_out, int out_size, void* d_ws, size_t ws_size,
                              hipStream_t stream) {
  const float* x         = (const float*)d_in[0];
  const float* edge_attr = (const float*)d_in[1];
  const int*   edge_idx  = (const int*)d_in[2];
  // d_in[3] = batch: unused — every gathered vn row equals vn_emb[0]
  const float* feat_W = (const float*)d_in[4];
  const float* feat_b = (const float*)d_in[5];
  const float* vn_emb = (const float*)d_in[6];
  const float* edge_W = (const float*)d_in[7];
  const float* edge_b = (const float*)d_in[8];
  const float* eps    = (const float*)d_in[9];
  const float* W1     = (const float*)d_in[10];
  const float* b1     = (const float*)d_in[11];
  const float* bn1_g  = (const float*)d_in[12];
  const float* bn1_b  = (const float*)d_in[13];
  const float* bn1_m  = (const float*)d_in[14];
  const float* bn1_v  = (const float*)d_in[15];
  const float* W2     = (const float*)d_in[16];
  const float* b2     = (const float*)d_in[17];
  const float* bn_g   = (const float*)d_in[18];
  const float* bn_b   = (const float*)d_in[19];
  const float* bn_m   = (const float*)d_in[20];
  const float* bn_v   = (const float*)d_in[21];

  const int N = in_sizes[0] / IN_DIM;   // 50000 (multiple of 16)
  const int E = in_sizes[1] / IN_DIM;   // 600000 (multiple of 32)

  float* h0   = (float*)d_out;          // h0 lives in d_out, finalized in place
  float* aggr = (float*)d_ws;           // N*EMB floats

  // Packed weight scratch, after aggr
  const unsigned long long nelem = (unsigned long long)N * EMB;
  float* packF  = aggr  + nelem;                 // 128*128
  float* packE  = packF + IN_DIM * EMB;          // 128*128
  float* packW1 = packE + IN_DIM * EMB;          // 128*256
  float* packW2 = packW1 + EMB * HID;            // 256*128

  zero_f32_kernel<<<(unsigned)((nelem + 255) / 256), 256, 0, stream>>>(aggr, nelem);

  // Repack weights into fragment-major layout (elements = K*N each)
  pack_w_kernel<<<(IN_DIM * EMB / 2 + 255) / 256, 256, 0, stream>>>(feat_W, packF, IN_DIM, EMB);
  pack_w_kernel<<<(IN_DIM * EMB / 2 + 255) / 256, 256, 0, stream>>>(edge_W, packE, IN_DIM, EMB);
  pack_w_kernel<<<(EMB * HID / 2 + 255) / 256, 256, 0, stream>>>(W1, packW1, EMB, HID);
  pack_w_kernel<<<(HID * EMB / 2 + 255) / 256, 256, 0, stream>>>(W2, packW2, HID, EMB);

  encode_kernel<<<(N + 127) / 128, 256, 0, stream>>>(x, packF, feat_b, vn_emb, h0, N);

  edge_kernel<<<(E + 255) / 256, 256, 0, stream>>>(edge_attr, packE, edge_b,
                                                   edge_idx, edge_idx + E, h0, aggr, E);

  mlp_kernel<<<N / 16, 32, 0, stream>>>(h0, aggr, eps,
                                        packW1, b1, bn1_g, bn1_b, bn1_m, bn1_v,
                                        packW2, b2, bn_g, bn_b, bn_m, bn_v);
}